// Quantize_9517647527982
// MI455X (gfx1250) — compile-verified
//
#include <hip/hip_runtime.h>

// -----------------------------------------------------------------------------
// VQ-VAE quantize for MI455X (gfx1250, wave32, WMMA).
//
// Roofline: GEMM 131072x512x64 = 8.6 GFLOP dominates; memory ~70MB = ~3us at
// 23.3 TB/s. Use V_WMMA_F32_16X16X4_F32 for the -2*x@embed term; argmin does
// not need ||x||^2 (row-constant), only esq[n] - 2*dot. Codebook (128KB) lives
// in LDS (transposed, padded stride 66) so B fragments are ds_load_b64.
// -----------------------------------------------------------------------------

typedef float v2f __attribute__((ext_vector_type(2)));
typedef float v8f __attribute__((ext_vector_type(8)));

#define DECAY 0.99f
#define EPSF  1e-5f

constexpr int DIM   = 64;
constexpr int NEMB  = 512;
constexpr int NROWS = 256 * 512;        // 131072 input vectors
constexpr int ROWS_PER_BLOCK = 128;     // 8 waves * 16 rows
constexpr int ESTRIDE = 66;             // padded LDS stride (bank-conflict free)

__global__ __launch_bounds__(256) void vq_main(
    const float* __restrict__ inp,        // [NROWS, 64]
    const float* __restrict__ embed,      // [64, 512]
    float* __restrict__ out_quant,        // [NROWS, 64]
    float* __restrict__ out_ind,          // [NROWS] (indices as float)
    float* __restrict__ ws_counts,        // [512]
    float* __restrict__ ws_embed_sum,     // [64, 512]
    float* __restrict__ ws_sse)           // [1]
{
    __shared__ float sE[NEMB * ESTRIDE];  // embed transposed: sE[n*66 + k]
    __shared__ float sEsq[NEMB];          // ||embed[:,n]||^2

    const int tid = threadIdx.x;

    // ---- stage codebook into LDS, transposed, coalesced global reads ----
    for (int i = tid; i < DIM * NEMB; i += 256) {
        int k = i >> 9;            // 0..63
        int n = i & (NEMB - 1);    // 0..511
        sE[n * ESTRIDE + k] = embed[i];
    }
    __syncthreads();
    for (int n = tid; n < NEMB; n += 256) {
        float s = 0.f;
        #pragma unroll
        for (int k = 0; k < DIM; ++k) { float e = sE[n * ESTRIDE + k]; s += e * e; }
        sEsq[n] = s;
    }
    __syncthreads();

    const int lane = tid & 31;
    const int wave = tid >> 5;
    const int h    = lane >> 4;        // lane half (0/1)
    const int ln   = lane & 15;
    const int row0 = blockIdx.x * ROWS_PER_BLOCK + wave * 16;
    const int myrow = ln;              // A-matrix: M = lane % 16
    const int grow  = row0 + myrow;

    // ---- A fragments per ISA layout: lane half 0 holds K=4kb+{0,1},
    //      lane half 1 holds K=4kb+{2,3}; 16 fragments cover K=0..63 ----
    v2f af[16];
    const float* xrow = inp + (size_t)grow * DIM;
    #pragma unroll
    for (int kb = 0; kb < 16; ++kb)
        af[kb] = *(const v2f*)(xrow + 4 * kb + 2 * h);

    float minv[8];
    int   mini[8];
    #pragma unroll
    for (int i = 0; i < 8; ++i) { minv[i] = 3.4e38f; mini[i] = 0; }

    // ---- sweep 32 column tiles of 16 codebook entries each ----
    for (int ct = 0; ct < NEMB / 16; ++ct) {
        const int ncol = ct * 16 + ln;                 // this lane's column
        const float* bp = sE + ncol * ESTRIDE + 2 * h; // B frag base (8B aligned)
        v8f c = {0.f, 0.f, 0.f, 0.f, 0.f, 0.f, 0.f, 0.f};
        #pragma unroll
        for (int kb = 0; kb < 16; ++kb) {
            v2f b = *(const v2f*)(bp + 4 * kb);        // ds_load_b64
            c = __builtin_amdgcn_wmma_f32_16x16x4_f32(
                    false, af[kb], false, b, (short)0, c, false, false);
        }
        // D layout: VGPR i -> row (i + 8h), col = ncol for this lane.
        const float e = sEsq[ncol];
        #pragma unroll
        for (int i = 0; i < 8; ++i) {
            float score = e - 2.0f * c[i];             // dist minus row-const
            if (score < minv[i]) { minv[i] = score; mini[i] = ncol; }
        }
    }

    // ---- per-slot min across the 16 lanes of each half (row-local) ----
    #pragma unroll
    for (int i = 0; i < 8; ++i) {
        #pragma unroll
        for (int off = 1; off < 16; off <<= 1) {
            float ov = __shfl_xor(minv[i], off, 32);
            int   oi = __shfl_xor(mini[i], off, 32);
            if (ov < minv[i] || (ov == minv[i] && oi < mini[i])) {
                minv[i] = ov; mini[i] = oi;
            }
        }
    }
    // gather this lane's row index: rows 0..7 live in lower half slots,
    // rows 8..15 in upper half slots.
    int myidx = 0;
    #pragma unroll
    for (int s = 0; s < 8; ++s) {
        int lo = __shfl(mini[s], 0, 32);
        int hi = __shfl(mini[s], 16, 32);
        if (myrow == s)     myidx = lo;
        if (myrow == s + 8) myidx = hi;
    }

    // ---- outputs: index, cluster count, quantized row, embed_sum, sse ----
    if (h == 0) {
        out_ind[grow] = (float)myidx;
        atomicAdd(ws_counts + myidx, 1.0f);
    }

    float sse = 0.f;
    const float* qsrc = sE + myidx * ESTRIDE;
    float* qdst = out_quant + (size_t)grow * DIM;
    #pragma unroll
    for (int kb = 0; kb < 16; ++kb) {
        int k = 4 * kb + 2 * h;
        v2f q = *(const v2f*)(qsrc + k);               // ds_load_b64
        *(v2f*)(qdst + k) = q;                         // global_store_b64
        float d0 = q.x - af[kb].x, d1 = q.y - af[kb].y;
        sse += d0 * d0 + d1 * d1;
        atomicAdd(ws_embed_sum + (size_t)(k    ) * NEMB + myidx, af[kb].x);
        atomicAdd(ws_embed_sum + (size_t)(k + 1) * NEMB + myidx, af[kb].y);
    }
    #pragma unroll
    for (int off = 16; off >= 1; off >>= 1) sse += __shfl_xor(sse, off, 32);
    if (lane == 0) atomicAdd(ws_sse, sse);
}

__global__ __launch_bounds__(512) void vq_finalize(
    const float* __restrict__ cluster_size,  // [512]
    const float* __restrict__ embed_avg,     // [64, 512]
    const float* __restrict__ ws_counts,
    const float* __restrict__ ws_embed_sum,
    const float* __restrict__ ws_sse,
    float* __restrict__ out_diff,
    float* __restrict__ out_embed,
    float* __restrict__ out_cs,
    float* __restrict__ out_avg)
{
    __shared__ float red[NEMB];
    const int t = threadIdx.x;
    float cnew = DECAY * cluster_size[t] + (1.0f - DECAY) * ws_counts[t];
    out_cs[t] = cnew;
    red[t] = cnew;
    __syncthreads();
    for (int off = 256; off >= 1; off >>= 1) {
        if (t < off) red[t] += red[t + off];
        __syncthreads();
    }
    const float n  = red[0];
    const float cs = (cnew + EPSF) / (n + (float)NEMB * EPSF) * n;
    const float inv = 1.0f / cs;
    for (int k = 0; k < DIM; ++k) {
        int i = k * NEMB + t;
        float ea = DECAY * embed_avg[i] + (1.0f - DECAY) * ws_embed_sum[i];
        out_avg[i]   = ea;
        out_embed[i] = ea * inv;
    }
    if (t == 0)
        out_diff[0] = 1.25f * ws_sse[0] / (float)(NROWS * DIM);
}

extern "C" void kernel_launch(void* const* d_in, const int* in_sizes, int n_in,
                              void* d_out, int out_size, void* d_ws, size_t ws_size,
                              hipStream_t stream) {
    const float* inp          = (const float*)d_in[0];  // [256,512,64]
    const float* embed        = (const float*)d_in[1];  // [64,512]
    const float* cluster_size = (const float*)d_in[2];  // [512]
    const float* embed_avg    = (const float*)d_in[3];  // [64,512]

    float* out = (float*)d_out;
    float* out_quant = out;                         // 8388608
    float* out_diff  = out + 8388608;               // 1
    float* out_ind   = out + 8388609;               // 131072
    float* out_embed = out + 8388609 + 131072;      // 32768
    float* out_cs    = out + 8388609 + 131072 + 32768;       // 512
    float* out_avg   = out + 8388609 + 131072 + 32768 + 512; // 32768

    float* ws           = (float*)d_ws;
    float* ws_counts    = ws;              // 512
    float* ws_embed_sum = ws + 512;        // 32768
    float* ws_sse       = ws + 512 + 32768;// 1
    hipMemsetAsync(d_ws, 0, (512 + 32768 + 1) * sizeof(float), stream);

    vq_main<<<NROWS / ROWS_PER_BLOCK, 256, 0, stream>>>(
        inp, embed, out_quant, out_ind, ws_counts, ws_embed_sum, ws_sse);
    vq_finalize<<<1, 512, 0, stream>>>(
        cluster_size, embed_avg, ws_counts, ws_embed_sum, ws_sse,
        out_diff, out_embed, out_cs, out_avg);
}